// HybridQLSTMGraphQNN_65481071408608
// MI455X (gfx1250) — compile-verified
//
#include <hip/hip_runtime.h>
#include <math.h>

typedef __attribute__((ext_vector_type(2))) float v2f;
typedef __attribute__((ext_vector_type(8))) float v8f;

#define S_LEN  8192
#define E_DIM  1024
#define H_DIM  4
#define T_TAGS 50
#define W_COLS (E_DIM + H_DIM)   // 1028

// ---------------------------------------------------------------------------
// Phase 1: xpart[t][h] = dot(emb[sentence[t]], W_lin[h][0:1024])
// Tall-skinny GEMM via v_wmma_f32_16x16x4_f32 (full f32, bandwidth-bound).
// One wave per 16-row tile. B = W_lin^T padded N:4->16 with a 0/1 lane mask.
// K-slice remap: lanes 0-15 load global k..k+3, lanes 16-31 load k+4..k+7;
//   wmma0 uses (.x,.y) -> K-slice {k,k+1,k+4,k+5}
//   wmma1 uses (.z,.w) -> K-slice {k+2,k+3,k+6,k+7}
// A and B share the mapping, so the K-sum is unchanged.
// ---------------------------------------------------------------------------
__global__ __launch_bounds__(32) void qlstm_xproj_wmma(
    const int*   __restrict__ sentence,
    const float* __restrict__ emb,
    const float* __restrict__ Wl,
    float*       __restrict__ xpart)
{
  const int tile = blockIdx.x;         // 512 tiles of 16 rows
  const int lane = threadIdx.x;        // 0..31 (wave32)
  const int m    = lane & 15;          // A row-in-tile / B column N
  const int kSel = (lane >> 4) << 2;   // 0 or 4: half of each 8-K chunk

  const int tok = sentence[tile * 16 + m];
  const float* arow = emb + (size_t)tok * E_DIM + kSel;
  const int    bRow = (m < H_DIM) ? m : 0;          // safe row for padded lanes
  const float  bMsk = (m < H_DIM) ? 1.0f : 0.0f;    // zero out padded columns
  const float* brow = Wl + (size_t)bRow * W_COLS + kSel;

  v8f acc0 = {0.f,0.f,0.f,0.f,0.f,0.f,0.f,0.f};
  v8f acc1 = {0.f,0.f,0.f,0.f,0.f,0.f,0.f,0.f};
  for (int k = 0; k < E_DIM; k += 8) {
    const float4 av = *(const float4*)(arow + k);
    const float4 bv = *(const float4*)(brow + k);
    v2f a0, a1, b0, b1;
    a0.x = av.x;        a0.y = av.y;
    a1.x = av.z;        a1.y = av.w;
    b0.x = bv.x * bMsk; b0.y = bv.y * bMsk;
    b1.x = bv.z * bMsk; b1.y = bv.w * bMsk;
    acc0 = __builtin_amdgcn_wmma_f32_16x16x4_f32(false, a0, false, b0,
                                                 (short)0, acc0, false, false);
    acc1 = __builtin_amdgcn_wmma_f32_16x16x4_f32(false, a1, false, b1,
                                                 (short)0, acc1, false, false);
  }
  v8f acc = acc0 + acc1;

  // D layout: lanes 0-15 hold N=lane, M=0..7 in acc[0..7]; lanes 16-31 M=8..15
  float* outp = xpart + (size_t)(tile * 16) * H_DIM;
  if (lane < H_DIM) {
#pragma unroll
    for (int v = 0; v < 8; ++v) outp[v * H_DIM + lane] = acc[v];
  } else if (lane >= 16 && lane < 16 + H_DIM) {
#pragma unroll
    for (int v = 0; v < 8; ++v) outp[(8 + v) * H_DIM + (lane - 16)] = acc[v];
  }
}

// ---------------------------------------------------------------------------
// Phase 2: sequential LSTM scan, single wave, lane-parallel quantum layer.
// Circuit closed form (RX layer -> CNOT ring is a basis permutation):
//   z_w = cos(gate_w + theta_w);
//   outputs = [z1*z2*z3, z0*z1, z0*z1*z2, z0*z1*z2*z3]
// Lane = gate*4 + wire (gates: 0=f sig, 1=i sig, 2=g tanh, 3=o sig).
// ---------------------------------------------------------------------------
__device__ __forceinline__ float sig_fast(float x) {
  return 1.0f / (1.0f + __expf(-x));
}

__global__ __launch_bounds__(32) void qlstm_scan(
    const float* __restrict__ xpart,
    const float* __restrict__ Wl,
    const float* __restrict__ b_lin,
    const float* __restrict__ th_f,
    const float* __restrict__ th_i,
    const float* __restrict__ th_g,
    const float* __restrict__ th_o,
    float*       __restrict__ houts)
{
  extern __shared__ float s_x[];                // 8192*4 floats = 128 KB LDS
  const int lane = threadIdx.x;

  // Stage xpart into LDS (short-latency reads inside the serial loop)
  for (int i = lane * 4; i < S_LEN * H_DIM; i += 32 * 4)
    *(float4*)(s_x + i) = *(const float4*)(xpart + i);
  __syncthreads();

  const int q = (lane >> 2) & 3;                // gate id
  const int w = lane & 3;                       // wire id
  const float4 wh = *(const float4*)(Wl + (size_t)w * W_COLS + E_DIM); // Whh row w
  const float  bb = b_lin[w];
  const float  th = (q == 0) ? th_f[w] : (q == 1) ? th_i[w]
                  : (q == 2) ? th_g[w] : th_o[w];
  // branchless sigmoid/tanh: tanh(x) = 2*sigmoid(2x) - 1
  const float ascl = (q == 2) ? 2.0f : 1.0f;
  const float amul = (q == 2) ? 2.0f : 1.0f;
  const float aadd = (q == 2) ? -1.0f : 0.0f;

  float h0 = 0.f, h1 = 0.f, h2 = 0.f, h3 = 0.f;
  float c  = 0.f;                               // cell state for wire w (replicated)
  float xv = s_x[w];
  const int base = lane & 12;                   // canonical group in lanes 0..15

  for (int t = 0; t < S_LEN; ++t) {
    float xn = (t + 1 < S_LEN) ? s_x[(t + 1) * H_DIM + w] : 0.f;  // prefetch

    float gv = xv + bb;
    gv = fmaf(wh.x, h0, gv); gv = fmaf(wh.y, h1, gv);
    gv = fmaf(wh.z, h2, gv); gv = fmaf(wh.w, h3, gv);
    float z = __cosf(gv + th);

    float z0 = __shfl(z, base | 0, 32);
    float z1 = __shfl(z, base | 1, 32);
    float z2 = __shfl(z, base | 2, 32);
    float z3 = __shfl(z, base | 3, 32);
    float p01 = z0 * z1, p012 = p01 * z2, p0123 = p012 * z3;
    float p123 = (z1 * z2) * z3;
    float e = (w == 0) ? p123 : (w == 1) ? p01 : (w == 2) ? p012 : p0123;

    float a = fmaf(amul, sig_fast(ascl * e), aadd);  // sigmoid or tanh per gate

    float fg = __shfl(a,  0 + w, 32);
    float ig = __shfl(a,  4 + w, 32);
    float gg = __shfl(a,  8 + w, 32);
    float og = __shfl(a, 12 + w, 32);
    float cn = fmaf(fg, c, ig * gg);
    float hn = og * (fmaf(2.0f, sig_fast(2.0f * cn), -1.0f));
    c = cn;

    if (lane < H_DIM) houts[t * H_DIM + lane] = hn;  // fire-and-forget store
    h0 = __shfl(hn, 0, 32); h1 = __shfl(hn, 1, 32);
    h2 = __shfl(hn, 2, 32); h3 = __shfl(hn, 3, 32);
    xv = xn;
  }
}

// ---------------------------------------------------------------------------
// Phase 3: logits = h @ W_tag^T + b_tag, then log_softmax over 50 tags.
// ---------------------------------------------------------------------------
__global__ __launch_bounds__(256) void qlstm_tag(
    const float* __restrict__ houts,
    const float* __restrict__ Wt,
    const float* __restrict__ bt,
    float*       __restrict__ out)
{
  const int t = blockIdx.x * blockDim.x + threadIdx.x;
  if (t >= S_LEN) return;
  const float4 h = *(const float4*)(houts + (size_t)t * H_DIM);

  float l[T_TAGS];
  float mx = -INFINITY;
#pragma unroll
  for (int k = 0; k < T_TAGS; ++k) {
    const float4 wv = *(const float4*)(Wt + k * H_DIM);
    float v = bt[k];
    v = fmaf(wv.x, h.x, v); v = fmaf(wv.y, h.y, v);
    v = fmaf(wv.z, h.z, v); v = fmaf(wv.w, h.w, v);
    l[k] = v;
    mx = fmaxf(mx, v);
  }
  float s = 0.f;
#pragma unroll
  for (int k = 0; k < T_TAGS; ++k) s += __expf(l[k] - mx);
  const float lg = mx + __logf(s);
  float* op = out + (size_t)t * T_TAGS;
#pragma unroll
  for (int k = 0; k < T_TAGS; ++k) op[k] = l[k] - lg;
}

// ---------------------------------------------------------------------------
extern "C" void kernel_launch(void* const* d_in, const int* in_sizes, int n_in,
                              void* d_out, int out_size, void* d_ws, size_t ws_size,
                              hipStream_t stream) {
  const int*   sentence = (const int*)  d_in[0];
  const float* emb      = (const float*)d_in[1];
  const float* Wl       = (const float*)d_in[2];
  const float* bl       = (const float*)d_in[3];
  const float* thf      = (const float*)d_in[4];
  const float* thi      = (const float*)d_in[5];
  const float* thg      = (const float*)d_in[6];
  const float* tho      = (const float*)d_in[7];
  const float* Wt       = (const float*)d_in[8];
  const float* bt       = (const float*)d_in[9];
  float* out   = (float*)d_out;
  float* xpart = (float*)d_ws;                       // [8192*4]
  float* houts = xpart + S_LEN * H_DIM;              // [8192*4]

  qlstm_xproj_wmma<<<S_LEN / 16, 32, 0, stream>>>(sentence, emb, Wl, xpart);
  qlstm_scan<<<1, 32, S_LEN * H_DIM * sizeof(float), stream>>>(
      xpart, Wl, bl, thf, thi, thg, tho, houts);
  qlstm_tag<<<(S_LEN + 255) / 256, 256, 0, stream>>>(houts, Wt, bt, out);
}